// UniformShardedEmbeddingBags_35673998360772
// MI455X (gfx1250) — compile-verified
//
#include <hip/hip_runtime.h>
#include <hip/hip_bf16.h>
#include <stdint.h>

// UniformShardedEmbeddingBags for MI455X (gfx1250).
// weights: [E=200000, T=16, D=64] f32 ; indices: [T*B*20] ; offsets: [T*B+1]
// out[b, t, :] = sum over bag (t*B + b) of weights[idx, t, :]
//
// Memory-bound random gather (~363 MB -> ~15.6 us @ 23.3 TB/s).
// One wave32 per bag. CDNA5 async global->LDS gather: each
// global_load_async_to_lds_b128 issue moves 2 full rows (512 B) with no
// VGPR destinations (ASYNCcnt-tracked). Index loads are issued
// speculatively (uniform pooling) in parallel with the offsets load to
// shorten the dependent-latency chain, and the LDS column sum is
// overlapped with the in-flight tail via a split s_wait_asynccnt.

#define T_TABLES        16
#define EMBED_D         64
#define POOL_MAX        20
#define WAVES_PER_BLOCK 8
#define THREADS         (WAVES_PER_BLOCK * 32)

__device__ __forceinline__ void async_gather_2rows(uint64_t wbase, int rowIdx,
                                                   uint32_t tByte, uint32_t part,
                                                   uint32_t ldsWaveByte, uint32_t r)
{
    // byte offset: idx * (T*D*4) + t * (D*4) + part*16   (< 2^31, fits u32)
    const uint32_t gOff   = (uint32_t)rowIdx * (T_TABLES * EMBED_D * 4)
                            + tByte + part * 16u;
    const uint64_t gaddr  = wbase + gOff;
    const uint32_t ldsOff = ldsWaveByte + r * (EMBED_D * 4) + part * 16u;
    asm volatile("global_load_async_to_lds_b128 %0, %1, off"
                 :: "v"(ldsOff), "v"(gaddr)
                 : "memory");
}

__global__ __launch_bounds__(THREADS)
void embedding_bag_async_kernel(const float* __restrict__ weights,
                                const int*   __restrict__ indices,
                                const int*   __restrict__ offsets,
                                float*       __restrict__ out,
                                int numBags, int bShift, int nIdx)
{
    // 8 waves * 20 rows * 64 f32 = 40 KB staging (of 320 KB per WGP)
    __shared__ float lds[WAVES_PER_BLOCK * POOL_MAX * EMBED_D];

    const int lane        = threadIdx.x & 31;
    const int waveInBlock = threadIdx.x >> 5;
    const int g           = blockIdx.x * WAVES_PER_BLOCK + waveInBlock;
    if (g >= numBags) return;

    const int t = g >> bShift;             // table id (bags grouped by table)
    const int b = g & ((1 << bShift) - 1); // batch id within table

    // Speculative index load (uniform pooling => start == g*POOL) issued
    // CONCURRENTLY with the offsets load: one fewer dependent HBM trip.
    const int startSpec = g * POOL_MAX;
    int iAddr = startSpec + lane;
    if (iAddr >= nIdx) iAddr = nIdx - 1;   // clamp: inactive lanes of last bag
    const int myIdxSpec = indices[iAddr];

    const int start = offsets[g];
    int n = offsets[g + 1] - start;
    if (n > POOL_MAX) n = POOL_MAX;
    if (n < 0) n = 0;

    const uint32_t ldsWaveByte = (uint32_t)waveInBlock * (POOL_MAX * EMBED_D * 4);
    const uint32_t tByte       = (uint32_t)t * (EMBED_D * 4);
    const uint64_t wbase       = (uint64_t)(uintptr_t)weights;

    const uint32_t half = (uint32_t)(lane >> 4); // which of the 2 rows per issue
    const uint32_t part = (uint32_t)(lane & 15); // 16-byte chunk within a row

    float2 acc = make_float2(0.0f, 0.0f);
    const float* wrow = lds + (size_t)waveInBlock * (POOL_MAX * EMBED_D);

    if (n == POOL_MAX && start == startSpec) {
        // ---- fast path: uniform pooling, speculation verified ----
        // 10 back-to-back async issues (2 rows / 512 B each); all 32 lanes
        // active every issue; ASYNCcnt rides at 10 per wave.
        #pragma unroll
        for (uint32_t j = 0; j < POOL_MAX / 2; ++j) {
            const uint32_t r      = 2u * j + half;
            const int      rowIdx = __shfl(myIdxSpec, (int)r, 32);
            async_gather_2rows(wbase, rowIdx, tByte, part, ldsWaveByte, r);
        }

        // Async loads complete in order: cnt<=5 => issues 0..4 (rows 0..9)
        // are resident. Sum them while the tail is still in flight.
        asm volatile("s_wait_asynccnt 0x5" ::: "memory");
        #pragma unroll
        for (int r = 0; r < POOL_MAX / 2; ++r) {
            const float2 v = *(const float2*)(wrow + r * EMBED_D + lane * 2);
            acc.x += v.x;
            acc.y += v.y;
        }

        asm volatile("s_wait_asynccnt 0x0" ::: "memory");
        #pragma unroll
        for (int r = POOL_MAX / 2; r < POOL_MAX; ++r) {
            const float2 v = *(const float2*)(wrow + r * EMBED_D + lane * 2);
            acc.x += v.x;
            acc.y += v.y;
        }
    } else {
        // ---- generic path: variable bag length / non-uniform offsets ----
        const int myIdx = (lane < n) ? indices[start + lane] : 0;

        const int nIssues = (n + 1) >> 1;
        for (int j = 0; j < nIssues; ++j) {
            const uint32_t r      = 2u * (uint32_t)j + half;
            const int      rowIdx = __shfl(myIdx, (int)r, 32);
            if ((int)r < n) {
                async_gather_2rows(wbase, rowIdx, tByte, part, ldsWaveByte, r);
            }
        }

        asm volatile("s_wait_asynccnt 0x0" ::: "memory");

        for (int r = 0; r < n; ++r) {
            const float2 v = *(const float2*)(wrow + r * EMBED_D + lane * 2);
            acc.x += v.x;
            acc.y += v.y;
        }
    }

    // out[b][t][:] — 256 B contiguous per wave, coalesced b64 stores.
    float2* dst = (float2*)(out + ((size_t)b * T_TABLES + t) * EMBED_D);
    dst[lane] = acc;
}

extern "C" void kernel_launch(void* const* d_in, const int* in_sizes, int n_in,
                              void* d_out, int out_size, void* d_ws, size_t ws_size,
                              hipStream_t stream)
{
    const float* weights = (const float*)d_in[0];
    const int*   sf      = (const int*)d_in[1];
    const int*   off     = (const int*)d_in[2];
    float*       out     = (float*)d_out;

    const int numBags = in_sizes[2] - 1;          // T * B
    const int nIdx    = in_sizes[1];              // T * B * POOL
    const int B       = numBags / T_TABLES;       // 4096 (power of two)
    int bShift = 0;
    while ((1 << bShift) < B) ++bShift;           // log2(B)

    const int blocks  = (numBags + WAVES_PER_BLOCK - 1) / WAVES_PER_BLOCK;

    embedding_bag_async_kernel<<<blocks, THREADS, 0, stream>>>(
        weights, sf, off, out, numBags, bShift, nIdx);
}